// CrossEntropyLoss_7799660609794
// MI455X (gfx1250) — compile-verified
//
#include <hip/hip_runtime.h>
#include <hip/hip_bf16.h>

typedef __attribute__((ext_vector_type(2))) float v2f;
typedef __attribute__((ext_vector_type(4))) float v4f;
typedef __attribute__((ext_vector_type(8))) float v8f;
typedef __attribute__((ext_vector_type(4))) int   v4i;

// Problem geometry (J=6, B=16, C=2, H=W=512)
#define CE_HW      262144u          // 512*512
#define CE_PLANES  96u              // J*B
#define CE_NPIX    25165824u        // 96 * HW
#define CE_GROUPS  6291456u         // NPIX / 4 (float4 groups; HW % 4 == 0 so no plane crossing)
#define CE_BLOCKS  2048
#define CE_THREADS 256

// Stage 1: streaming NLL partial sums. Pure bandwidth kernel: three 128-bit
// non-temporal loads per 4 pixels, v_exp/v_log for the logsumexp, fixed-tree
// wave32 + LDS reduction, one deterministic partial per block.
__global__ void ce_partial_kernel(const float* __restrict__ logits,
                                  const int*   __restrict__ target,
                                  float*       __restrict__ partial) {
    unsigned tid    = blockIdx.x * blockDim.x + threadIdx.x;
    unsigned stride = gridDim.x * blockDim.x;

    float acc = 0.0f;
    for (unsigned g = tid; g < CE_GROUPS; g += stride) {
        unsigned p0    = g << 2;                 // first pixel of the group
        unsigned plane = p0 >> 18;               // / HW   (0..95) = j*16 + b
        unsigned pix   = p0 & (CE_HW - 1u);      // % HW
        unsigned base  = plane * (2u * CE_HW) + pix;   // class-0 offset
        unsigned tbase = (plane & 15u) * CE_HW + pix;  // target offset (b,h,w)

        v4f l0 = __builtin_nontemporal_load((const v4f*)(logits + base));
        v4f l1 = __builtin_nontemporal_load((const v4f*)(logits + base + CE_HW));
        v4i tg = __builtin_nontemporal_load((const v4i*)(target + tbase));

#pragma unroll
        for (int k = 0; k < 4; ++k) {
            float x  = l0[k];
            float y  = l1[k];
            float mx = fmaxf(x, y);
            float lse = mx + __logf(__expf(x - mx) + __expf(y - mx));
            float picked = (tg[k] != 0) ? y : x;
            acc += lse - picked;
        }
    }

    // wave32 xor-tree reduction
#pragma unroll
    for (int m = 16; m > 0; m >>= 1)
        acc += __shfl_xor(acc, m, 32);

    __shared__ float smem[CE_THREADS / 32];
    int lane = threadIdx.x & 31;
    int wave = threadIdx.x >> 5;
    if (lane == 0) smem[wave] = acc;
    __syncthreads();

    if (threadIdx.x == 0) {
        float v = 0.0f;
#pragma unroll
        for (int i = 0; i < CE_THREADS / 32; ++i) v += smem[i];
        partial[blockIdx.x] = v;
    }
}

// Stage 2: single wave32. Each lane folds the 2048 block partials into two
// f32 accumulators (64 live values across the wave), which are packed as the
// B matrix of V_WMMA_F32_16X16X4_F32 with A = ones(16x4):
//   D[m][n] = sum_k B[k][n]   (column sums, full f32 precision)
// Row m=0 of D lives in D-VGPR0 lanes 0..15 (one column sum per lane);
// 4 xor-shuffles give the grand total in lane 0. Deterministic fixed tree.
__global__ void ce_final_kernel(const float* __restrict__ partial,
                                int nPartial,
                                float* __restrict__ out,
                                float invN) {
    int lane = threadIdx.x;              // 0..31, EXEC all ones (no divergence)

    float accA = 0.0f, accB = 0.0f;
    for (int i = lane; i < nPartial; i += 64)      accA += partial[i];
    for (int i = lane + 32; i < nPartial; i += 64) accB += partial[i];

    v2f a;  a.x = 1.0f; a.y = 1.0f;      // A = ones(16x4)
    v2f b;  b.x = accA; b.y = accB;      // B = 4x16 carrying the 64 partials
    v8f c = {};                          // C = 0

    v8f d = __builtin_amdgcn_wmma_f32_16x16x4_f32(
        /*neg_a=*/false, a, /*neg_b=*/false, b,
        /*c_mod=*/(short)0, c, /*reuse_a=*/false, /*reuse_b=*/false);

    float col = d[0];                    // lane n (0..15): column-sum over k
#pragma unroll
    for (int m = 8; m > 0; m >>= 1)      // reduce lanes 0..15 among themselves
        col += __shfl_xor(col, m, 32);

    if (lane == 0) out[0] = col * invN;
}

extern "C" void kernel_launch(void* const* d_in, const int* in_sizes, int n_in,
                              void* d_out, int out_size, void* d_ws, size_t ws_size,
                              hipStream_t stream) {
    const float* logits = (const float*)d_in[0];
    const int*   target = (const int*)d_in[1];
    float*       out    = (float*)d_out;
    float*       part   = (float*)d_ws;   // 2048 floats = 8 KB scratch

    ce_partial_kernel<<<CE_BLOCKS, CE_THREADS, 0, stream>>>(logits, target, part);
    ce_final_kernel<<<1, 32, 0, stream>>>(part, CE_BLOCKS, out,
                                          1.0f / (float)CE_NPIX);
}